// DynamicLocalGlobalRouter_10926396801635
// MI455X (gfx1250) — compile-verified
//
#include <hip/hip_runtime.h>
#include <hip/hip_bf16.h>

#define B_ 2
#define H_ 16
#define S_ 2048
#define D_ 64
#define NEGV (-1000000000.0f)

typedef float v2f __attribute__((ext_vector_type(2)));
typedef float v8f __attribute__((ext_vector_type(8)));

// LDS strides chosen so 16-lane b64 fragment reads are bank-conflict-free:
// stride % 64 == 4  ->  banks (4*r + 2*hi + {0,1}) all distinct across 32 lanes.
#define SQ_STRIDE  68   // 32 rows  (queries x dims)
#define SK_STRIDE  68   // 64 rows  (keys    x dims)
#define SVT_STRIDE 68   // 64 rows  (dims    x keys)  == V transposed
#define SP_STRIDE  52   // 16 rows  (queries x 48 keys), 52%64=52 -> 4*(13r%16) distinct

// ---------------- gate kernel: gate[bh] = sigmoid(mean_s(q) . w_gate + b) ----
__global__ __launch_bounds__(256)
void gate_kernel(const float* __restrict__ q, const float* __restrict__ wg,
                 const float* __restrict__ bg, float* __restrict__ gate) {
    __shared__ float red[8];
    const int bh = blockIdx.x;
    const float* qp = q + (size_t)bh * S_ * D_;
    float partial = 0.0f;
    for (int idx = threadIdx.x; idx < S_ * D_; idx += 256)
        partial += qp[idx] * wg[idx & (D_ - 1)];
    #pragma unroll
    for (int off = 1; off < 32; off <<= 1)
        partial += __shfl_xor(partial, off, 32);
    if ((threadIdx.x & 31) == 0) red[threadIdx.x >> 5] = partial;
    __syncthreads();
    if (threadIdx.x == 0) {
        float s = 0.0f;
        #pragma unroll
        for (int i = 0; i < 8; ++i) s += red[i];
        float x = s * (1.0f / (float)S_) + bg[0];
        gate[bh] = 1.0f / (1.0f + __expf(-x));
    }
}

// ---------------- local attention + blend --------------------------------
// grid: (S/32, B*H), block: 64 threads = 2 waves; wave w handles queries
// [g0+16w, g0+16w+16). Shared K/V window covers keys [g0-16, g0+48).
__global__ __launch_bounds__(64)
void attn_kernel(const float* __restrict__ q, const float* __restrict__ k,
                 const float* __restrict__ v, const float* __restrict__ gout,
                 const float* __restrict__ gate, float* __restrict__ out) {
    __shared__ float sQ [32 * SQ_STRIDE];
    __shared__ float sK [64 * SK_STRIDE];
    __shared__ float sVt[64 * SVT_STRIDE];          // [dim][key]
    __shared__ float sP [2 * 16 * SP_STRIDE];       // per-wave probs 16x48

    const int bh = blockIdx.y;
    const int g0 = blockIdx.x * 32;
    const size_t base = (size_t)bh * S_ * D_;
    const int tid = threadIdx.x;

    // ---- stage Q (scaled), K, V^T through LDS (coalesced global reads) ----
    #pragma unroll 4
    for (int idx = tid; idx < 32 * D_; idx += 64) {
        int r = idx >> 6, d = idx & 63;
        sQ[r * SQ_STRIDE + d] = q[base + (size_t)(g0 + r) * D_ + d] * 0.125f;
    }
    #pragma unroll 4
    for (int idx = tid; idx < 64 * D_; idx += 64) {
        int r = idx >> 6, d = idx & 63;
        int j  = g0 - 16 + r;
        int jc = min(max(j, 0), S_ - 1);
        size_t off = base + (size_t)jc * D_ + d;
        sK [r * SK_STRIDE  + d] = k[off];
        sVt[d * SVT_STRIDE + r] = v[off];
    }
    __syncthreads();

    const int lane = tid & 31;
    const int wv   = tid >> 5;          // wave's query tile (0..1)
    const int lr   = lane & 15;
    const int hi   = lane >> 4;

    // ---- scores: 3 key tiles x 16 k-steps of v_wmma_f32_16x16x4_f32 ------
    v8f acc0 = {}, acc1 = {}, acc2 = {};
    const float* qrow  = &sQ[(16 * wv + lr) * SQ_STRIDE + 2 * hi];
    const float* krow0 = &sK[(16 * wv + lr) * SK_STRIDE + 2 * hi];
    const float* krow1 = krow0 + 16 * SK_STRIDE;
    const float* krow2 = krow0 + 32 * SK_STRIDE;
    #pragma unroll
    for (int kk = 0; kk < 16; ++kk) {
        v2f a  = *(const v2f*)(qrow  + 4 * kk);
        v2f b0 = *(const v2f*)(krow0 + 4 * kk);
        v2f b1 = *(const v2f*)(krow1 + 4 * kk);
        v2f b2 = *(const v2f*)(krow2 + 4 * kk);
        acc0 = __builtin_amdgcn_wmma_f32_16x16x4_f32(false, a, false, b0, (short)0, acc0, false, false);
        acc1 = __builtin_amdgcn_wmma_f32_16x16x4_f32(false, a, false, b1, (short)0, acc1, false, false);
        acc2 = __builtin_amdgcn_wmma_f32_16x16x4_f32(false, a, false, b2, (short)0, acc2, false, false);
    }

    // ---- band/boundary mask + row softmax (16-lane xor reductions) -------
    // D-layout: lane holds key col n = lr; rows m = 8*hi + g in acc[g].
    const int j0 = g0 + 16 * wv - 16 + lr;      // absolute key, tile 0
    #pragma unroll
    for (int g = 0; g < 8; ++g) {
        int m  = 8 * hi + g;
        int sq = g0 + 16 * wv + m;
        float s0 = acc0[g], s1 = acc1[g], s2 = acc2[g];
        int r0 = j0 - sq, r1 = r0 + 16, r2 = r0 + 32;
        bool ok0 = (r0 >= -16) && (r0 <= 16) && (j0 >= 0)      && (j0 < S_);
        bool ok1 = (r1 >= -16) && (r1 <= 16) && (j0 + 16 >= 0) && (j0 + 16 < S_);
        bool ok2 = (r2 >= -16) && (r2 <= 16) && (j0 + 32 >= 0) && (j0 + 32 < S_);
        s0 = ok0 ? s0 : NEGV; s1 = ok1 ? s1 : NEGV; s2 = ok2 ? s2 : NEGV;
        float mx = fmaxf(fmaxf(s0, s1), s2);
        #pragma unroll
        for (int off = 1; off < 16; off <<= 1)
            mx = fmaxf(mx, __shfl_xor(mx, off, 32));
        float e0 = __expf(s0 - mx), e1 = __expf(s1 - mx), e2 = __expf(s2 - mx);
        float sum = e0 + e1 + e2;
        #pragma unroll
        for (int off = 1; off < 16; off <<= 1)
            sum += __shfl_xor(sum, off, 32);
        float inv = 1.0f / sum;
        float* pr = &sP[(wv * 16 + m) * SP_STRIDE + lr];
        pr[0]  = e0 * inv;
        pr[16] = e1 * inv;
        pr[32] = e2 * inv;
    }
    __syncthreads();

    // ---- out = P(16x48) x V(48x64): 4 D-chunks x 12 k-steps --------------
    v8f o0 = {}, o1 = {}, o2 = {}, o3 = {};
    const float* prow = &sP[(wv * 16 + lr) * SP_STRIDE + 2 * hi];
    const float* vb0  = &sVt[(lr +  0) * SVT_STRIDE + 16 * wv + 2 * hi];
    const float* vb1  = &sVt[(lr + 16) * SVT_STRIDE + 16 * wv + 2 * hi];
    const float* vb2  = &sVt[(lr + 32) * SVT_STRIDE + 16 * wv + 2 * hi];
    const float* vb3  = &sVt[(lr + 48) * SVT_STRIDE + 16 * wv + 2 * hi];
    #pragma unroll
    for (int kk = 0; kk < 12; ++kk) {
        v2f a  = *(const v2f*)(prow + 4 * kk);
        v2f b0 = *(const v2f*)(vb0  + 4 * kk);
        v2f b1 = *(const v2f*)(vb1  + 4 * kk);
        v2f b2 = *(const v2f*)(vb2  + 4 * kk);
        v2f b3 = *(const v2f*)(vb3  + 4 * kk);
        o0 = __builtin_amdgcn_wmma_f32_16x16x4_f32(false, a, false, b0, (short)0, o0, false, false);
        o1 = __builtin_amdgcn_wmma_f32_16x16x4_f32(false, a, false, b1, (short)0, o1, false, false);
        o2 = __builtin_amdgcn_wmma_f32_16x16x4_f32(false, a, false, b2, (short)0, o2, false, false);
        o3 = __builtin_amdgcn_wmma_f32_16x16x4_f32(false, a, false, b3, (short)0, o3, false, false);
    }

    // ---- blend with gate and global_out ----------------------------------
    const float gt = gate[bh];
    const float og = 1.0f - gt;
    v8f oc[4] = {o0, o1, o2, o3};
    #pragma unroll
    for (int c = 0; c < 4; ++c) {
        #pragma unroll
        for (int g = 0; g < 8; ++g) {
            int m = 8 * hi + g;
            size_t idx = base + (size_t)(g0 + 16 * wv + m) * D_ + 16 * c + lr;
            out[idx] = gt * oc[c][g] + og * gout[idx];
        }
    }
}

extern "C" void kernel_launch(void* const* d_in, const int* in_sizes, int n_in,
                              void* d_out, int out_size, void* d_ws, size_t ws_size,
                              hipStream_t stream) {
    const float* q    = (const float*)d_in[0];
    const float* k    = (const float*)d_in[1];
    const float* v    = (const float*)d_in[2];
    const float* gout = (const float*)d_in[3];
    // d_in[4] = buckets (unused by the reference computation)
    const float* wg   = (const float*)d_in[5];
    const float* bg   = (const float*)d_in[6];
    float* out  = (float*)d_out;
    float* gate = (float*)d_ws;                      // 32 floats

    gate_kernel<<<B_ * H_, 256, 0, stream>>>(q, wg, bg, gate);
    attn_kernel<<<dim3(S_ / 32, B_ * H_), 64, 0, stream>>>(q, k, v, gout, gate, out);
}